// MultiHeadLatentAttention_5480378270013
// MI455X (gfx1250) — compile-verified
//
#include <hip/hip_runtime.h>
#include <hip/hip_bf16.h>
#include <math.h>

// ---------------------------------------------------------------------------
// MLA forward for MI455X (gfx1250), fp32 end-to-end using V_WMMA_F32_16X16X4_F32
// GEMM: 32(M) x 64(N) register tile per wave, 8 accumulators, K-step 4.
// ---------------------------------------------------------------------------

#define BB 2
#define SS 1024
#define DM 2048
#define NH 16
#define KVL 512
#define QL 512
#define DN 128
#define DR 64
#define DV 128

typedef __attribute__((ext_vector_type(2))) float v2f;
typedef __attribute__((ext_vector_type(8))) float v8f;

#define WMMA_F32(a, b, c) \
    __builtin_amdgcn_wmma_f32_16x16x4_f32(false, (a), false, (b), (short)0, (c), false, false)

// Generic strided, optionally batched (z = zb*Hdiv + zh) fp32 WMMA GEMM:
//   C[m,n] = sum_k A[m,k]*B[k,n] (+ bias[n]) (+ addmat[m,n])
// A element (m,k): A + m*sA_row + k                (k contiguous, sA_row even)
// B element (k,n): B + k*sB_k + n*sB_n             (BCONT => sB_k == 1)
// C element (m,n): C + m*sC_row + n
// Per-batch offsets: zb*s*_zb + zh*s*_zh  (addmat is never batched).
// One wave computes a 32x64 tile (2 M-subtiles x 4 N-subtiles of 16x16).
// Requires: M % 32 == 0, N % 64 == 0, K % 4 == 0.
template <bool BCONT>
__global__ void __launch_bounds__(256)
wmma_gemm_f32(const float* __restrict__ A, int sA_row, long sA_zb, long sA_zh,
              const float* __restrict__ B, int sB_k, int sB_n, long sB_zb, long sB_zh,
              float* __restrict__ C, int sC_row, long sC_zb, long sC_zh,
              const float* __restrict__ bias,
              const float* __restrict__ addmat, int sAdd_row,
              int Hdiv, int N, int K)
{
    const int wave = threadIdx.x >> 5;
    const int lane = threadIdx.x & 31;
    const int tn0 = (blockIdx.x * 8 + wave) * 64;   // first column of this wave's tile
    if (tn0 >= N) return;                           // wave-uniform exit (before WMMA)
    const int tm0 = blockIdx.y * 32;                // first row of this wave's tile
    const int zb = blockIdx.z / Hdiv;
    const int zh = blockIdx.z % Hdiv;
    A += zb * sA_zb + zh * sA_zh;
    B += zb * sB_zb + zh * sB_zh;
    C += zb * sC_zb + zh * sC_zh;

    // A 16x4 layout: lane = M (0..15) in each half; VGPR v holds K = 2*(lane>>4)+v
    // B 4x16 layout: lane = N (0..15) in each half; VGPR v holds K = 2*(lane>>4)+v
    const int ml = lane & 15;
    const int kh = (lane >> 4) << 1;

    const float* __restrict__ ApL = A + (long)(tm0 + ml) * sA_row + kh;
    const float* __restrict__ ApH = ApL + (long)16 * sA_row;
    const float* __restrict__ Bp0 = B + (long)(tn0 + ml) * sB_n + (long)kh * sB_k;
    const float* __restrict__ Bp1 = Bp0 + (long)16 * sB_n;
    const float* __restrict__ Bp2 = Bp0 + (long)32 * sB_n;
    const float* __restrict__ Bp3 = Bp0 + (long)48 * sB_n;
    const long sBk4 = 4L * sB_k;

    v8f aL0 = {}, aL1 = {}, aL2 = {}, aL3 = {};
    v8f aH0 = {}, aH1 = {}, aH2 = {}, aH3 = {};

#pragma unroll 2
    for (int k = 0; k < K; k += 4) {
        v2f aL, aH, b0, b1, b2, b3;
        aL = *(const v2f*)ApL;                      // b64: k-contiguous, 8B aligned
        aH = *(const v2f*)ApH;
        if (BCONT) {                                // sB_k == 1: contiguous pair
            b0 = *(const v2f*)Bp0;
            b1 = *(const v2f*)Bp1;
            b2 = *(const v2f*)Bp2;
            b3 = *(const v2f*)Bp3;
        } else {
            b0.x = Bp0[0]; b0.y = Bp0[sB_k];
            b1.x = Bp1[0]; b1.y = Bp1[sB_k];
            b2.x = Bp2[0]; b2.y = Bp2[sB_k];
            b3.x = Bp3[0]; b3.y = Bp3[sB_k];
        }
        ApL += 4; ApH += 4;
        Bp0 += sBk4; Bp1 += sBk4; Bp2 += sBk4; Bp3 += sBk4;

        aL0 = WMMA_F32(aL, b0, aL0);
        aL1 = WMMA_F32(aL, b1, aL1);
        aL2 = WMMA_F32(aL, b2, aL2);
        aL3 = WMMA_F32(aL, b3, aL3);
        aH0 = WMMA_F32(aH, b0, aH0);
        aH1 = WMMA_F32(aH, b1, aH1);
        aH2 = WMMA_F32(aH, b2, aH2);
        aH3 = WMMA_F32(aH, b3, aH3);
    }

    // C/D layout: element acc[v] is (M = 8*(lane>>4) + v, N = lane&15)
    const int mL = tm0 + ((lane >> 4) << 3);   // low 16-row subtile base
    v8f accL[4] = { aL0, aL1, aL2, aL3 };
    v8f accH[4] = { aH0, aH1, aH2, aH3 };
#pragma unroll
    for (int j = 0; j < 4; ++j) {
        const int nn = tn0 + j * 16 + ml;
        const float bv = bias ? bias[nn] : 0.0f;
#pragma unroll
        for (int v = 0; v < 8; ++v) {
            float valL = accL[j][v] + bv;
            float valH = accH[j][v] + bv;
            if (addmat) {
                valL += addmat[(long)(mL + v) * sAdd_row + nn];
                valH += addmat[(long)(mL + 16 + v) * sAdd_row + nn];
            }
            C[(long)(mL + v) * sC_row + nn] = valL;
            C[(long)(mL + 16 + v) * sC_row + nn] = valH;
        }
    }
}

// LayerNorm over 512 columns; one block (256 threads) per row.
__global__ void __launch_bounds__(256)
layernorm512(const float* __restrict__ src, int src_stride,
             float* __restrict__ dst, int dst_stride,
             const float* __restrict__ scale, const float* __restrict__ bias)
{
    const int r = blockIdx.x;
    const int tid = threadIdx.x;
    const float* p = src + (long)r * src_stride;
    float a = p[tid];
    float b = p[tid + 256];

    __shared__ float red[256];
    red[tid] = a + b;
    __syncthreads();
    for (int o = 128; o > 0; o >>= 1) {
        if (tid < o) red[tid] += red[tid + o];
        __syncthreads();
    }
    const float mu = red[0] * (1.0f / 512.0f);
    __syncthreads();

    const float da = a - mu, db = b - mu;
    red[tid] = da * da + db * db;
    __syncthreads();
    for (int o = 128; o > 0; o >>= 1) {
        if (tid < o) red[tid] += red[tid + o];
        __syncthreads();
    }
    const float inv = rsqrtf(red[0] * (1.0f / 512.0f) + 1e-6f);

    float* q = dst + (long)r * dst_stride;
    q[tid]       = da * inv * scale[tid] + bias[tid];
    q[tid + 256] = db * inv * scale[tid + 256] + bias[tid + 256];
}

// Build per-head contiguous Q' (scaled, roped) and K' (roped) of width 192.
// One block per (b,s).
__global__ void __launch_bounds__(256)
pack_qk(const float* __restrict__ q_proj,     // (B*S, NH*192)
        const float* __restrict__ kv_proj,    // (B*S, NH*256)
        const float* __restrict__ kv_lat,     // (B*S, 576): [512 latent | 64 k_rope_in]
        float* __restrict__ Qh,               // (B,NH,S,192)
        float* __restrict__ Kh)               // (B,NH,S,192)
{
    const int bs = blockIdx.x;                // 0 .. B*S-1
    const int b = bs >> 10;
    const int s = bs & (SS - 1);
    const int tid = threadIdx.x;

    __shared__ float sinv[DR / 2], cosv[DR / 2], krope[DR];
    if (tid < DR / 2) {
        float freq = powf(10000.0f, -(2.0f * (float)tid) / (float)DR);
        float ang = (float)s * freq;
        sinv[tid] = sinf(ang);
        cosv[tid] = cosf(ang);
    }
    __syncthreads();
    if (tid < DR / 2) {
        const float* kr = kv_lat + (long)bs * (KVL + DR) + KVL;
        float xe = kr[2 * tid], xo = kr[2 * tid + 1];
        krope[2 * tid]     = xe * cosv[tid] - xo * sinv[tid];
        krope[2 * tid + 1] = xe * sinv[tid] + xo * cosv[tid];
    }
    __syncthreads();

    const float qn_scale = 0.08838834764831845f; // 1/sqrt(128)
    const float qr_scale = 0.125f;               // 1/sqrt(64)

    for (int h = 0; h < NH; ++h) {
        float* qdst = Qh + (((long)(b * NH + h) * SS + s) * (DN + DR));
        float* kdst = Kh + (((long)(b * NH + h) * SS + s) * (DN + DR));
        const float* qsrc = q_proj + ((long)bs * (NH * (DN + DR)) + h * (DN + DR));
        const float* ksrc = kv_proj + ((long)bs * (NH * (DN + DV)) + h * (DN + DV));
        if (tid < DN) {
            qdst[tid] = qsrc[tid] * qn_scale;
            kdst[tid] = ksrc[tid];
        } else if (tid < DN + DR / 2) {
            int i = tid - DN;                  // rope pair index 0..31
            float xe = qsrc[DN + 2 * i], xo = qsrc[DN + 2 * i + 1];
            qdst[DN + 2 * i]     = (xe * cosv[i] - xo * sinv[i]) * qr_scale;
            qdst[DN + 2 * i + 1] = (xe * sinv[i] + xo * cosv[i]) * qr_scale;
            kdst[DN + 2 * i]     = krope[2 * i];
            kdst[DN + 2 * i + 1] = krope[2 * i + 1];
        }
    }
}

// In-place softmax over rows of length 1024; one block (256 threads) per row.
__global__ void __launch_bounds__(256)
softmax1024(float* __restrict__ data)
{
    float* p = data + (long)blockIdx.x * SS;
    const int tid = threadIdx.x;
    float v0 = p[tid], v1 = p[tid + 256], v2 = p[tid + 512], v3 = p[tid + 768];

    __shared__ float red[256];
    red[tid] = fmaxf(fmaxf(v0, v1), fmaxf(v2, v3));
    __syncthreads();
    for (int o = 128; o > 0; o >>= 1) {
        if (tid < o) red[tid] = fmaxf(red[tid], red[tid + o]);
        __syncthreads();
    }
    const float m = red[0];
    __syncthreads();

    v0 = expf(v0 - m); v1 = expf(v1 - m); v2 = expf(v2 - m); v3 = expf(v3 - m);
    red[tid] = v0 + v1 + v2 + v3;
    __syncthreads();
    for (int o = 128; o > 0; o >>= 1) {
        if (tid < o) red[tid] += red[tid + o];
        __syncthreads();
    }
    const float inv = 1.0f / red[0];
    p[tid] = v0 * inv; p[tid + 256] = v1 * inv;
    p[tid + 512] = v2 * inv; p[tid + 768] = v3 * inv;
}

extern "C" void kernel_launch(void* const* d_in, const int* in_sizes, int n_in,
                              void* d_out, int out_size, void* d_ws, size_t ws_size,
                              hipStream_t stream) {
    const float* x             = (const float*)d_in[0];
    const float* mask          = (const float*)d_in[1];
    const float* wkv_a_w       = (const float*)d_in[2];
    const float* wkv_a_b       = (const float*)d_in[3];
    const float* kv_norm_scale = (const float*)d_in[4];
    const float* kv_norm_bias  = (const float*)d_in[5];
    const float* wkv_b_w       = (const float*)d_in[6];
    const float* wkv_b_b       = (const float*)d_in[7];
    const float* wq_a_w        = (const float*)d_in[8];
    const float* wq_a_b        = (const float*)d_in[9];
    const float* q_norm_scale  = (const float*)d_in[10];
    const float* q_norm_bias   = (const float*)d_in[11];
    const float* wq_b_w        = (const float*)d_in[12];
    const float* wq_b_b        = (const float*)d_in[13];
    const float* wo_w          = (const float*)d_in[14];
    const float* wo_b          = (const float*)d_in[15];

    float* out  = (float*)d_out;                       // (B*S, 2048)
    float* attn = out + (long)BB * SS * DM;            // (B, S, H, S)

    float* ws = (float*)d_ws;
    float* kv_lat  = ws; ws += (long)BB * SS * (KVL + DR);        // (2048, 576)
    float* ln_kv   = ws; ws += (long)BB * SS * KVL;               // (2048, 512)
    float* qdown   = ws; ws += (long)BB * SS * QL;                // (2048, 512)
    float* kv_proj = ws; ws += (long)BB * SS * NH * (DN + DV);    // (2048, 4096)
    float* q_proj  = ws; ws += (long)BB * SS * NH * (DN + DR);    // (2048, 3072)
    float* Qh      = ws; ws += (long)BB * NH * SS * (DN + DR);    // (32, 1024, 192)
    float* Kh      = ws; ws += (long)BB * NH * SS * (DN + DR);    // (32, 1024, 192)
    float* hout    = ws; ws += (long)BB * SS * NH * DV;           // (2048, 2048)

    const dim3 blk(256);
    const int M = BB * SS;  // 2048
    // One block covers 32 rows x 512 columns.
    #define GEMM_GRID(MM, NN) dim3(((NN) + 511) / 512, (MM) / 32, 1)
    #define GEMM_GRID_B(MM, NN, ZZ) dim3(((NN) + 511) / 512, (MM) / 32, ZZ)

    // 1) kv_lat = x @ wkv_a_w + wkv_a_b            (2048 x 576, K=2048)
    wmma_gemm_f32<false><<<GEMM_GRID(M, KVL + DR), blk, 0, stream>>>(
        x, DM, 0, 0,
        wkv_a_w, KVL + DR, 1, 0, 0,
        kv_lat, KVL + DR, 0, 0,
        wkv_a_b, nullptr, 0,
        1, KVL + DR, DM);

    // 2) qdown = x @ wq_a_w + wq_a_b               (2048 x 512, K=2048)
    wmma_gemm_f32<false><<<GEMM_GRID(M, QL), blk, 0, stream>>>(
        x, DM, 0, 0,
        wq_a_w, QL, 1, 0, 0,
        qdown, QL, 0, 0,
        wq_a_b, nullptr, 0,
        1, QL, DM);

    // 3) LayerNorm kv latent (cols 0..511 of kv_lat) -> ln_kv
    layernorm512<<<M, 256, 0, stream>>>(kv_lat, KVL + DR, ln_kv, KVL,
                                        kv_norm_scale, kv_norm_bias);
    // 4) LayerNorm qdown in place
    layernorm512<<<M, 256, 0, stream>>>(qdown, QL, qdown, QL,
                                        q_norm_scale, q_norm_bias);

    // 5) kv_proj = ln_kv @ wkv_b_w + wkv_b_b       (2048 x 4096, K=512)
    wmma_gemm_f32<false><<<GEMM_GRID(M, NH * (DN + DV)), blk, 0, stream>>>(
        ln_kv, KVL, 0, 0,
        wkv_b_w, NH * (DN + DV), 1, 0, 0,
        kv_proj, NH * (DN + DV), 0, 0,
        wkv_b_b, nullptr, 0,
        1, NH * (DN + DV), KVL);

    // 6) q_proj = qdown @ wq_b_w + wq_b_b          (2048 x 3072, K=512)
    wmma_gemm_f32<false><<<GEMM_GRID(M, NH * (DN + DR)), blk, 0, stream>>>(
        qdown, QL, 0, 0,
        wq_b_w, NH * (DN + DR), 1, 0, 0,
        q_proj, NH * (DN + DR), 0, 0,
        wq_b_b, nullptr, 0,
        1, NH * (DN + DR), QL);

    // 7) Pack Q'/K' per head (RoPE + score prescale folded in)
    pack_qk<<<M, 256, 0, stream>>>(q_proj, kv_proj, kv_lat, Qh, Kh);

    // 8) scores = Q' @ K'^T + mask  -> attn[b,s,h,t]   (32 batches of 1024x1024, K=192)
    //    B is K' transposed: sB_k == 1 -> contiguous b64 B loads.
    wmma_gemm_f32<true><<<GEMM_GRID_B(SS, SS, BB * NH), blk, 0, stream>>>(
        Qh, DN + DR, (long)NH * SS * (DN + DR), (long)SS * (DN + DR),
        Kh, 1, DN + DR, (long)NH * SS * (DN + DR), (long)SS * (DN + DR),
        attn, NH * SS, (long)SS * NH * SS, (long)SS,
        nullptr, mask, SS,
        NH, SS, DN + DR);

    // 9) softmax over t, in place in output attn buffer
    softmax1024<<<BB * SS * NH, 256, 0, stream>>>(attn);

    // 10) hout[b,s,h*128+d] = attn[b,s,h,:] @ v[b,:,h,:]   (32 batches of 1024x128, K=1024)
    wmma_gemm_f32<false><<<GEMM_GRID_B(SS, DV, BB * NH), blk, 0, stream>>>(
        attn, NH * SS, (long)SS * NH * SS, (long)SS,
        kv_proj + DN, NH * (DN + DV), 1, (long)SS * NH * (DN + DV), (long)(DN + DV),
        hout, NH * DV, (long)SS * NH * DV, (long)DV,
        nullptr, nullptr, 0,
        NH, DV, SS);

    // 11) out = hout @ wo_w + wo_b                 (2048 x 2048, K=2048)
    wmma_gemm_f32<false><<<GEMM_GRID(M, DM), blk, 0, stream>>>(
        hout, NH * DV, 0, 0,
        wo_w, DM, 1, 0, 0,
        out, DM, 0, 0,
        wo_b, nullptr, 0,
        1, DM, DM);
}